// ContentBasedAttention_48893907698051
// MI455X (gfx1250) — compile-verified
//
#include <hip/hip_runtime.h>
#include <hip/hip_bf16.h>
#include <math.h>

typedef float v2f __attribute__((ext_vector_type(2)));
typedef float v4f __attribute__((ext_vector_type(4)));
typedef float v8f __attribute__((ext_vector_type(8)));

#define B_DIM 8
#define D_DIM 256
#define E_DIM 512
#define F_DIM 256

// Native gfx1250 V_TANH_F32 (single TRANS op, co-executes with VALU).
#if defined(__has_builtin)
#if __has_builtin(__builtin_amdgcn_tanhf)
#define FAST_TANH(x) __builtin_amdgcn_tanhf(x)
#endif
#endif
#ifndef FAST_TANH
#define FAST_TANH(x) tanhf(x)
#endif

// ---------------------------------------------------------------------------
// FC GEMM: Y[b,i,j] = sum_f X[b,i,f] * W[j,f] + bias[j]
// X: (B, R, F_DIM), W: (F_DIM, F_DIM) row-major.
// One wave computes one 16x16 output tile with V_WMMA_F32_16X16X4_F32.
// A-matrix (16x4 f32) per-lane layout: M = lane%16, K pair = 2*(lane/16)+{0,1}
// B-matrix (4x16 f32) per-lane layout: N = lane%16, K pair = 2*(lane/16)+{0,1}
// Both are contiguous float2 loads (W^T row == W row along f).
// ---------------------------------------------------------------------------
template <int R>
__global__ __launch_bounds__(256) void fc_wmma_kernel(
    const float* __restrict__ X, const float* __restrict__ W,
    const float* __restrict__ bias, float* __restrict__ Y) {
  const int lane = threadIdx.x & 31;
  const int wave = threadIdx.x >> 5;
  const int tilesM = R / 16, tilesN = F_DIM / 16;

  int tile = blockIdx.x * 8 + wave;
  int b = tile / (tilesM * tilesN);
  int rem = tile % (tilesM * tilesN);
  int row0 = (rem / tilesN) * 16;
  int col0 = (rem % tilesN) * 16;

  const int mn = lane & 15;          // M for A, N for B
  const int khi = (lane >> 4) << 1;  // 0 or 2

  const float* Xrow = X + ((size_t)b * R + row0 + mn) * F_DIM + khi;
  const float* Wrow = W + (size_t)(col0 + mn) * F_DIM + khi;

  v8f acc = {};
#pragma unroll 4
  for (int k = 0; k < F_DIM; k += 4) {
    v2f a = *(const v2f*)(Xrow + k);
    v2f w = *(const v2f*)(Wrow + k);
    acc = __builtin_amdgcn_wmma_f32_16x16x4_f32(false, a, false, w, (short)0,
                                                acc, false, false);
  }

  // C/D layout: VGPR r -> M = r (lanes 0-15) or 8+r (lanes 16-31); N = lane%16
  const int mbase = row0 + ((lane >> 4) << 3);
  const float bj = bias[col0 + mn];
  float* Yp = Y + (size_t)b * R * F_DIM + col0 + mn;
#pragma unroll
  for (int r = 0; r < 8; ++r)
    Yp[(size_t)(mbase + r) * F_DIM] = acc[r] + bj;
}

// ---------------------------------------------------------------------------
// Fused scores + softmax: one block (256 threads, 8 waves) per (b, d).
// scores[e] = sum_f v[f] * tanh(fc_dec[b,d,f] + fc_enc[b,e,f])
// Each wave owns 64 contiguous e. Lane L owns the contiguous f-chunk
// [8L, 8L+8): two global_load_b128 per e, v[f] and fc_dec[f] hoisted into
// registers, native v_tanh_f32 in the hot loop, shfl-xor wave reduction.
// Writes normalized attention weights straight to the output buffer.
// ---------------------------------------------------------------------------
__global__ __launch_bounds__(256) void score_softmax_kernel(
    const float* __restrict__ fc_dec, const float* __restrict__ fc_enc,
    const float* __restrict__ v, float* __restrict__ attn) {
  __shared__ __align__(16) float ssc[E_DIM];
  __shared__ float sred[8];

  const int tid = threadIdx.x;
  const int bd = blockIdx.x;  // b * D + d
  const int b = bd / D_DIM;
  const int lane = tid & 31;
  const int wave = tid >> 5;

  // Hoist per-lane loop invariants: this lane's 8 contiguous f values.
  const float* decrow = fc_dec + (size_t)bd * F_DIM + lane * 8;
  const v4f dlo = *(const v4f*)(decrow);
  const v4f dhi = *(const v4f*)(decrow + 4);
  const v4f vlo = *(const v4f*)(v + lane * 8);
  const v4f vhi = *(const v4f*)(v + lane * 8 + 4);

  const float* encb = fc_enc + (size_t)b * E_DIM * F_DIM + lane * 8;
#pragma unroll 2
  for (int ei = 0; ei < E_DIM / 8; ++ei) {
    const int e = wave * (E_DIM / 8) + ei;
    const float* er = encb + (size_t)e * F_DIM;
    const v4f elo = *(const v4f*)(er);
    const v4f ehi = *(const v4f*)(er + 4);
    float acc = 0.f;
#pragma unroll
    for (int i = 0; i < 4; ++i) acc += vlo[i] * FAST_TANH(dlo[i] + elo[i]);
#pragma unroll
    for (int i = 0; i < 4; ++i) acc += vhi[i] * FAST_TANH(dhi[i] + ehi[i]);
#pragma unroll
    for (int off = 16; off > 0; off >>= 1) acc += __shfl_xor(acc, off, 32);
    if (lane == 0) ssc[e] = acc;
  }
  __syncthreads();

  // Softmax over E=512 (each thread owns 2 scores).
  const float s0 = ssc[tid];
  const float s1 = ssc[tid + 256];
  float m = fmaxf(s0, s1);
#pragma unroll
  for (int off = 16; off > 0; off >>= 1) m = fmaxf(m, __shfl_xor(m, off, 32));
  if (lane == 0) sred[wave] = m;
  __syncthreads();
  if (tid == 0) {
    float mm = sred[0];
    for (int i = 1; i < 8; ++i) mm = fmaxf(mm, sred[i]);
    sred[0] = mm;
  }
  __syncthreads();
  const float gmax = sred[0];

  const float p0 = __expf(s0 - gmax);
  const float p1 = __expf(s1 - gmax);
  float sum = p0 + p1;
#pragma unroll
  for (int off = 16; off > 0; off >>= 1) sum += __shfl_xor(sum, off, 32);
  __syncthreads();  // everyone has read sred[0] before it is overwritten
  if (lane == 0) sred[wave] = sum;
  __syncthreads();
  if (tid == 0) {
    float ss = 0.f;
    for (int i = 0; i < 8; ++i) ss += sred[i];
    sred[0] = ss;
  }
  __syncthreads();
  const float inv = 1.0f / sred[0];

  float* ap = attn + (size_t)bd * E_DIM;
  ap[tid] = p0 * inv;
  ap[tid + 256] = p1 * inv;
}

// ---------------------------------------------------------------------------
// context[b,d,j] = sum_e attn[b,d,e] * enc[b,e,j]   — batched GEMM, K = 512.
// Same WMMA tiling; B operand rows of enc are strided (two scalar loads).
// ---------------------------------------------------------------------------
__global__ __launch_bounds__(256) void ctx_wmma_kernel(
    const float* __restrict__ attn, const float* __restrict__ enc,
    float* __restrict__ ctx) {
  const int lane = threadIdx.x & 31;
  const int wave = threadIdx.x >> 5;
  const int tilesM = D_DIM / 16, tilesN = F_DIM / 16;

  int tile = blockIdx.x * 8 + wave;
  int b = tile / (tilesM * tilesN);
  int rem = tile % (tilesM * tilesN);
  int row0 = (rem / tilesN) * 16;
  int col0 = (rem % tilesN) * 16;

  const int mn = lane & 15;
  const int khi = (lane >> 4) << 1;

  const float* Arow = attn + ((size_t)b * D_DIM + row0 + mn) * E_DIM + khi;
  const float* Bcol = enc + ((size_t)b * E_DIM + khi) * F_DIM + col0 + mn;

  v8f acc = {};
#pragma unroll 4
  for (int k = 0; k < E_DIM; k += 4) {
    v2f a = *(const v2f*)(Arow + k);
    v2f w;
    w.x = Bcol[(size_t)k * F_DIM];
    w.y = Bcol[(size_t)(k + 1) * F_DIM];
    acc = __builtin_amdgcn_wmma_f32_16x16x4_f32(false, a, false, w, (short)0,
                                                acc, false, false);
  }

  const int mbase = row0 + ((lane >> 4) << 3);
  float* Cp = ctx + (size_t)b * D_DIM * F_DIM + col0 + mn;
#pragma unroll
  for (int r = 0; r < 8; ++r)
    Cp[(size_t)(mbase + r) * F_DIM] = acc[r];
}

extern "C" void kernel_launch(void* const* d_in, const int* in_sizes, int n_in,
                              void* d_out, int out_size, void* d_ws,
                              size_t ws_size, hipStream_t stream) {
  const float* dec = (const float*)d_in[0];    // (B, D, F)
  const float* enc = (const float*)d_in[1];    // (B, E, F)
  const float* W_enc = (const float*)d_in[2];  // (F, F)
  const float* b_enc = (const float*)d_in[3];  // (F,)
  const float* W_dec = (const float*)d_in[4];  // (F, F)
  const float* b_dec = (const float*)d_in[5];  // (F,)
  const float* v = (const float*)d_in[6];      // (F,)

  float* ctx_out = (float*)d_out;                             // (B, D, F)
  float* attn_out = ctx_out + (size_t)B_DIM * D_DIM * F_DIM;  // (B, D, E)

  float* fc_enc = (float*)d_ws;                            // (B, E, F)
  float* fc_dec = fc_enc + (size_t)B_DIM * E_DIM * F_DIM;  // (B, D, F)

  const int enc_tiles = B_DIM * (E_DIM / 16) * (F_DIM / 16);  // 4096
  const int dec_tiles = B_DIM * (D_DIM / 16) * (F_DIM / 16);  // 2048
  const int ctx_tiles = B_DIM * (D_DIM / 16) * (F_DIM / 16);  // 2048

  fc_wmma_kernel<E_DIM><<<enc_tiles / 8, 256, 0, stream>>>(enc, W_enc, b_enc,
                                                           fc_enc);
  fc_wmma_kernel<D_DIM><<<dec_tiles / 8, 256, 0, stream>>>(dec, W_dec, b_dec,
                                                           fc_dec);
  score_softmax_kernel<<<B_DIM * D_DIM, 256, 0, stream>>>(fc_dec, fc_enc, v,
                                                          attn_out);
  ctx_wmma_kernel<<<ctx_tiles / 8, 256, 0, stream>>>(attn_out, enc, ctx_out);
}